// VoxelGNN_D_32220844654632
// MI455X (gfx1250) — compile-verified
//
#include <hip/hip_runtime.h>
#include <hip/hip_bf16.h>
#include <math.h>
#include <stdint.h>

typedef float v2f __attribute__((ext_vector_type(2)));
typedef float v8f __attribute__((ext_vector_type(8)));

#define NB 16           // NUM_BATCH
#define DH 128          // H
#define D2 256          // 2*H

// GEMM tiling
#define KC 16           // K chunk staged in LDS per buffer
#define LDB_PAD 272     // 256 + 16: half-wave row pairs land on disjoint banks
#define LDA_PAD 20      // 16 + 4: conflict-free A-fragment reads

#define USE_ASYNC_LDS 1 // CDNA5 global_load_async_to_lds_* path

// ---------------------------------------------------------------------------
// Small utility kernels
// ---------------------------------------------------------------------------
__global__ void zero_f32_kernel(float* __restrict__ p, long long n) {
  long long i = (long long)blockIdx.x * blockDim.x + threadIdx.x;
  if (i < n) p[i] = 0.0f;
}

__global__ void zero_i32_kernel(int* __restrict__ p, long long n) {
  long long i = (long long)blockIdx.x * blockDim.x + threadIdx.x;
  if (i < n) p[i] = 0;
}

__global__ void init_base_kernel(int* __restrict__ base) {
  int i = threadIdx.x;
  if (i < NB) base[i] = 0x7FFFFFFF;
}

// segment_min(vfc, vbi) over NUM_BATCH segments
__global__ void seg_min_kernel(const int* __restrict__ vbi, const int* __restrict__ vfc,
                               int* __restrict__ base, int N) {
  int i = blockIdx.x * blockDim.x + threadIdx.x;
  if (i < N) atomicMin(&base[vbi[i]], vfc[i]);
}

// degree counts + relative positions for both edge sets
__global__ void edge_prep_kernel(const int* __restrict__ e1, const int* __restrict__ e2,
                                 const float* __restrict__ v,
                                 int* __restrict__ cnt1, int* __restrict__ cnt2,
                                 float* __restrict__ dpos1, float* __restrict__ dpos2, int E) {
  int i = blockIdx.x * blockDim.x + threadIdx.x;
  if (i >= 2 * E) return;
  if (i < E) {
    int s = e1[i], d = e1[E + i];
    atomicAdd(&cnt1[d], 1);
    #pragma unroll
    for (int j = 0; j < 3; ++j)
      dpos1[(size_t)i * 3 + j] = v[(size_t)d * 13 + 3 + j] - v[(size_t)s * 13 + 3 + j];
  } else {
    int k = i - E;
    int s = e2[k], d = e2[E + k];
    atomicAdd(&cnt2[d], 1);
    #pragma unroll
    for (int j = 0; j < 3; ++j)
      dpos2[(size_t)k * 3 + j] = v[(size_t)d * 13 + 3 + j] - v[(size_t)s * 13 + 3 + j];
  }
}

// x[:, :128] = feat @ W_fe + b_fe + PE[level];  x[:, 128:] = l @ W_le + b_le
__global__ void node_init_kernel(const float* __restrict__ v, const float* __restrict__ l,
                                 const int* __restrict__ vbi, const int* __restrict__ vfc,
                                 const int* __restrict__ base,
                                 const float* __restrict__ Wfe, const float* __restrict__ bfe,
                                 const float* __restrict__ Wle, const float* __restrict__ ble,
                                 float* __restrict__ x, int N) {
  int n = blockIdx.x;
  int c = threadIdx.x;
  if (n >= N) return;
  if (c < DH) {
    float acc = bfe[c];
    #pragma unroll
    for (int k = 0; k < 10; ++k) {
      float f = (k < 3) ? v[(size_t)n * 13 + k] : v[(size_t)n * 13 + k + 3];
      acc += f * Wfe[k * DH + c];
    }
    int level = vfc[n] - base[vbi[n]];
    int i2 = (c >> 1) * 2;
    float freq = __expf(-(float)i2 * (logf(10000.0f) / (float)DH));
    float ang = (float)level * freq;
    acc += (c & 1) ? __cosf(ang) : __sinf(ang);
    x[(size_t)n * D2 + c] = acc;
  } else {
    int c2 = c - DH;
    float acc = ble[c2];
    #pragma unroll
    for (int k = 0; k < 10; ++k) acc += l[(size_t)n * 10 + k] * Wle[k * DH + c2];
    x[(size_t)n * D2 + c] = acc;
  }
}

// ---------------------------------------------------------------------------
// Chunk stager: global -> LDS (async-to-LDS on CDNA5, ds_store fallback)
// B chunk: KC x 256 (padded rows), A chunk: 16 x KC (padded rows)
// ---------------------------------------------------------------------------
__device__ __forceinline__ void stage_chunk(const float* __restrict__ A,
                                            const float* __restrict__ B,
                                            float* sA, float* sB,
                                            int row0, int K, int kc, int tid) {
  // B: KC*256 = 4096 floats = 1024 float4 -> 4 per thread (256 threads)
  const float* Bg = B + (size_t)kc * KC * D2;
  #pragma unroll
  for (int i = 0; i < (KC * D2) / (256 * 4); ++i) {
    int idx = tid + i * 256;          // 0 .. 1023
    int r = idx >> 6;                 // 64 float4 per 256-wide row
    int c = (idx & 63) << 2;
    const float4* src = (const float4*)(Bg + (size_t)r * D2 + c);
    float* dst = &sB[r * LDB_PAD + c];
#if USE_ASYNC_LDS
    asm volatile("global_load_async_to_lds_b128 %0, %1, off"
                 :: "v"((unsigned)(uintptr_t)dst), "v"(src)
                 : "memory");
#else
    *(float4*)dst = *src;
#endif
  }
  // A: 16*KC = 256 floats = 64 float4 -> threads 0..63
  if (tid < (16 * KC) / 4) {
    int r = tid >> 2;                 // 4 float4 per KC=16 row
    int c = (tid & 3) << 2;
    const float4* src = (const float4*)(A + (size_t)(row0 + r) * K + (size_t)kc * KC + c);
    float* dst = &sA[r * LDA_PAD + c];
#if USE_ASYNC_LDS
    asm volatile("global_load_async_to_lds_b128 %0, %1, off"
                 :: "v"((unsigned)(uintptr_t)dst), "v"(src)
                 : "memory");
#else
    *(float4*)dst = *src;
#endif
  }
}

__device__ __forceinline__ void wait_async_lds() {
#if USE_ASYNC_LDS
  asm volatile("s_wait_asynccnt 0x0" ::: "memory");
#endif
}

// ---------------------------------------------------------------------------
// WMMA f32 GEMM: C[M x 256] (+)= A[M x K] @ B[K x 256] (+ bias)
// block = 8 waves covering 16 rows x 256 cols; wave w -> cols [32w, 32w+32)
// K staged through double-buffered LDS in KC-chunks; V_WMMA_F32_16X16X4_F32.
// M must be a multiple of 16; K a multiple of KC.
// ---------------------------------------------------------------------------
__global__ void __launch_bounds__(256)
gemm_wmma_f32_kernel(const float* __restrict__ A,
                     const float* __restrict__ B,
                     float* __restrict__ C,
                     const float* __restrict__ bias,
                     int M, int K, int accumulate) {
  __shared__ float sA[2][16 * LDA_PAD];   //  2.5 KB
  __shared__ float sB[2][KC * LDB_PAD];   // 34.0 KB

  const int row0 = blockIdx.x * 16;
  if (row0 >= M) return;
  const int tid  = threadIdx.x;
  const int wave = tid >> 5;
  const int lane = tid & 31;
  const int hi   = lane >> 4;        // lane group (0: lanes 0-15, 1: lanes 16-31)
  const int la   = lane & 15;
  const int col0 = wave * 32;
  const int colA = col0 + la;        // tile 0 column for this lane
  const int colB = col0 + 16 + la;   // tile 1 column for this lane

  v8f acc0, acc1;
  #pragma unroll
  for (int r = 0; r < 8; ++r) {
    int rr = row0 + r + hi * 8;      // C/D layout: VGPR r holds M=r (lo) / M=r+8 (hi)
    float c0 = accumulate ? C[(size_t)rr * D2 + colA] : 0.0f;
    float c1 = accumulate ? C[(size_t)rr * D2 + colB] : 0.0f;
    if (bias) { c0 += bias[colA]; c1 += bias[colB]; }
    acc0[r] = c0;
    acc1[r] = c1;
  }

  const int nchunk = K / KC;
  stage_chunk(A, B, sA[0], sB[0], row0, K, 0, tid);
  wait_async_lds();
  __syncthreads();

  for (int kc = 0; kc < nchunk; ++kc) {
    const int cur = kc & 1;
    if (kc + 1 < nchunk)
      stage_chunk(A, B, sA[cur ^ 1], sB[cur ^ 1], row0, K, kc + 1, tid);

    const float* sAc = sA[cur];
    const float* sBc = sB[cur];
    #pragma unroll
    for (int k = 0; k < KC; k += 4) {
      // A fragment (16x4): lane holds row la; VGPR0={K=k|K=k+2}, VGPR1={K=k+1|K=k+3}
      v2f a;
      const float* ap = sAc + la * LDA_PAD + k + hi * 2;
      a[0] = ap[0];
      a[1] = ap[1];
      // B fragments (4x16), row-striped across lanes
      v2f b0, b1;
      const float* bp = sBc + (k + hi * 2) * LDB_PAD;
      b0[0] = bp[colA];
      b0[1] = bp[LDB_PAD + colA];
      b1[0] = bp[colB];
      b1[1] = bp[LDB_PAD + colB];
      acc0 = __builtin_amdgcn_wmma_f32_16x16x4_f32(false, a, false, b0, (short)0, acc0,
                                                   false, false);
      acc1 = __builtin_amdgcn_wmma_f32_16x16x4_f32(false, a, false, b1, (short)0, acc1,
                                                   false, false);
    }
    wait_async_lds();   // next chunk's async transfers have landed
    __syncthreads();    // everyone done reading current buffer
  }

  #pragma unroll
  for (int r = 0; r < 8; ++r) {
    int rr = row0 + r + hi * 8;
    C[(size_t)rr * D2 + colA] = acc0[r];
    C[(size_t)rr * D2 + colB] = acc1[r];
  }
}

// ---------------------------------------------------------------------------
// Edge message + scatter: aggr[dst] += (YA[dst] + YB[src] + dpos@Wp + bm) * mask
// one thread per (edge, 4-channel group)
// ---------------------------------------------------------------------------
__global__ void edge_msg_kernel(const int* __restrict__ sd,   // src [0,E), dst [E,2E)
                                const float* __restrict__ dpos,
                                const float* __restrict__ mask,  // may be nullptr
                                const float* __restrict__ YA, const float* __restrict__ YB,
                                const float* __restrict__ Wp,    // 3 x 256
                                const float* __restrict__ bm,    // 256
                                float* __restrict__ aggr, int E) {
  long long t = (long long)blockIdx.x * blockDim.x + threadIdx.x;
  if (t >= (long long)E * 64) return;
  int eidx = (int)(t >> 6);
  int cg = ((int)t & 63) * 4;
  int s = sd[eidx], d = sd[E + eidx];
  float m = mask ? mask[eidx] : 1.0f;
  float dp0 = dpos[(size_t)eidx * 3 + 0];
  float dp1 = dpos[(size_t)eidx * 3 + 1];
  float dp2 = dpos[(size_t)eidx * 3 + 2];
  const float4 ya  = *(const float4*)(YA + (size_t)d * D2 + cg);
  const float4 yb  = *(const float4*)(YB + (size_t)s * D2 + cg);
  const float4 wp0 = *(const float4*)(Wp + cg);
  const float4 wp1 = *(const float4*)(Wp + D2 + cg);
  const float4 wp2 = *(const float4*)(Wp + 2 * D2 + cg);
  const float4 bb  = *(const float4*)(bm + cg);
  float* out = aggr + (size_t)d * D2 + cg;
  atomicAdd(out + 0, (ya.x + yb.x + dp0 * wp0.x + dp1 * wp1.x + dp2 * wp2.x + bb.x) * m);
  atomicAdd(out + 1, (ya.y + yb.y + dp0 * wp0.y + dp1 * wp1.y + dp2 * wp2.y + bb.y) * m);
  atomicAdd(out + 2, (ya.z + yb.z + dp0 * wp0.z + dp1 * wp1.z + dp2 * wp2.z + bb.z) * m);
  atomicAdd(out + 3, (ya.w + yb.w + dp0 * wp0.w + dp1 * wp1.w + dp2 * wp2.w + bb.w) * m);
}

__global__ void finalize_aggr_kernel(float* __restrict__ aggr, const int* __restrict__ cnt,
                                     long long total) {
  long long i = (long long)blockIdx.x * blockDim.x + threadIdx.x;
  if (i >= total) return;
  int n = (int)(i >> 8);
  aggr[i] /= fmaxf(1.0f, (float)cnt[n]);
}

// x += leaky_relu(upd, 0.01)
__global__ void epilogue_kernel(float* __restrict__ x, const float* __restrict__ upd,
                                long long total) {
  long long i = (long long)blockIdx.x * blockDim.x + threadIdx.x;
  if (i >= total) return;
  float u = upd[i];
  x[i] += (u > 0.0f) ? u : 0.01f * u;
}

// ---------------------------------------------------------------------------
// Host-side orchestration
// ---------------------------------------------------------------------------
extern "C" void kernel_launch(void* const* d_in, const int* in_sizes, int n_in,
                              void* d_out, int out_size, void* d_ws, size_t ws_size,
                              hipStream_t stream) {
  const float* v      = (const float*)d_in[0];
  const float* l      = (const float*)d_in[1];
  const int*   e      = (const int*)  d_in[2];
  const float* e_mask = (const float*)d_in[3];
  const int*   vbi    = (const int*)  d_in[4];
  const int*   vfc    = (const int*)  d_in[5];
  const int*   e2     = (const int*)  d_in[7];
  const float* W_fe   = (const float*)d_in[8];
  const float* b_fe   = (const float*)d_in[9];
  const float* W_le   = (const float*)d_in[10];
  const float* b_le   = (const float*)d_in[11];
  const float* W_m    = (const float*)d_in[12];
  const float* b_m    = (const float*)d_in[13];
  const float* W_m2   = (const float*)d_in[14];
  const float* b_m2   = (const float*)d_in[15];
  const float* W_u    = (const float*)d_in[16];
  const float* b_u    = (const float*)d_in[17];

  const int N = in_sizes[4];        // vbi has N entries
  const int E = in_sizes[3];        // e_mask has E entries

  float* x = (float*)d_out;         // x lives in d_out, updated in place

  // workspace layout
  char* ws = (char*)d_ws;
  size_t off = 0;
  float* YA    = (float*)(ws + off); off += (size_t)N * D2 * sizeof(float);
  float* YB    = (float*)(ws + off); off += (size_t)N * D2 * sizeof(float);
  float* aggr  = (float*)(ws + off); off += (size_t)N * D2 * sizeof(float);
  float* aggr2 = (float*)(ws + off); off += (size_t)N * D2 * sizeof(float);
  float* upd   = (float*)(ws + off); off += (size_t)N * D2 * sizeof(float);
  float* dpos1 = (float*)(ws + off); off += (size_t)E * 3 * sizeof(float);
  float* dpos2 = (float*)(ws + off); off += (size_t)E * 3 * sizeof(float);
  int*   cnt1  = (int*)  (ws + off); off += (size_t)N * sizeof(int);
  int*   cnt2  = (int*)  (ws + off); off += (size_t)N * sizeof(int);
  int*   base  = (int*)  (ws + off); off += (size_t)NB * sizeof(int);

  const long long NC = (long long)N * D2;
  const int TB = 256;
  const int gNC  = (int)((NC + TB - 1) / TB);
  const int gN   = (N + TB - 1) / TB;
  const int g2E  = (2 * E + TB - 1) / TB;
  const int gEC  = (int)(((long long)E * 64 + TB - 1) / TB);
  const int gGemm = N / 16;   // N = 50000 = 3125 * 16

  // ---- setup ----
  init_base_kernel<<<1, NB, 0, stream>>>(base);
  seg_min_kernel<<<gN, TB, 0, stream>>>(vbi, vfc, base, N);
  zero_i32_kernel<<<gN, TB, 0, stream>>>(cnt1, N);
  zero_i32_kernel<<<gN, TB, 0, stream>>>(cnt2, N);
  edge_prep_kernel<<<g2E, TB, 0, stream>>>(e, e2, v, cnt1, cnt2, dpos1, dpos2, E);
  node_init_kernel<<<N, 256, 0, stream>>>(v, l, vbi, vfc, base,
                                          W_fe, b_fe, W_le, b_le, x, N);

  const float* Wm_d  = W_m;                 // rows [0,256)
  const float* Wm_s  = W_m  + 256 * D2;     // rows [256,512)
  const float* Wm_p  = W_m  + 512 * D2;     // rows [512,515)
  const float* Wm2_d = W_m2;
  const float* Wm2_s = W_m2 + 256 * D2;
  const float* Wm2_p = W_m2 + 512 * D2;
  const float* Wu_x  = W_u;                 // rows [0,256)
  const float* Wu_a  = W_u + 256 * D2;      // rows [256,512)
  const float* Wu_a2 = W_u + 512 * D2;      // rows [512,768)

  for (int layer = 0; layer < 3; ++layer) {
    // edge set 1: Y = x @ [Wm_dst | Wm_src], scatter messages, normalize
    gemm_wmma_f32_kernel<<<gGemm, 256, 0, stream>>>(x, Wm_d, YA, nullptr, N, D2, 0);
    gemm_wmma_f32_kernel<<<gGemm, 256, 0, stream>>>(x, Wm_s, YB, nullptr, N, D2, 0);
    zero_f32_kernel<<<gNC, TB, 0, stream>>>(aggr, NC);
    edge_msg_kernel<<<gEC, TB, 0, stream>>>(e, dpos1, e_mask, YA, YB, Wm_p, b_m,
                                            aggr, E);
    finalize_aggr_kernel<<<gNC, TB, 0, stream>>>(aggr, cnt1, NC);

    // edge set 2 (unmasked), reusing Y buffers
    gemm_wmma_f32_kernel<<<gGemm, 256, 0, stream>>>(x, Wm2_d, YA, nullptr, N, D2, 0);
    gemm_wmma_f32_kernel<<<gGemm, 256, 0, stream>>>(x, Wm2_s, YB, nullptr, N, D2, 0);
    zero_f32_kernel<<<gNC, TB, 0, stream>>>(aggr2, NC);
    edge_msg_kernel<<<gEC, TB, 0, stream>>>(e2, dpos2, nullptr, YA, YB, Wm2_p, b_m2,
                                            aggr2, E);
    finalize_aggr_kernel<<<gNC, TB, 0, stream>>>(aggr2, cnt2, NC);

    // update: upd = x@Wu_x + b_u + aggr@Wu_a + aggr2@Wu_a2;  x += lrelu(upd)
    gemm_wmma_f32_kernel<<<gGemm, 256, 0, stream>>>(x,     Wu_x,  upd, b_u,     N, D2, 0);
    gemm_wmma_f32_kernel<<<gGemm, 256, 0, stream>>>(aggr,  Wu_a,  upd, nullptr, N, D2, 1);
    gemm_wmma_f32_kernel<<<gGemm, 256, 0, stream>>>(aggr2, Wu_a2, upd, nullptr, N, D2, 1);
    epilogue_kernel<<<gNC, TB, 0, stream>>>(x, upd, NC);
  }

  (void)n_in; (void)out_size; (void)ws_size; (void)off;
}